// MyModel_87522843560991
// MI455X (gfx1250) — compile-verified
//
#include <hip/hip_runtime.h>

// COO gather + scatter-add:  out[row[i], :] += mat[col[i], :]   (D = 128, f32)
//
// CDNA5 strategy: the row gather mat[col[i]] is done by the Tensor Data Mover
// in gather mode (tensor_load_to_lds, 32-bit indices, 8 rows / 4KB per op),
// double-buffered per wave and tracked on TENSORcnt. The wave overlaps the
// DMA with draining 4x global_atomic_add_f32 per lane into out (no-return,
// STOREcnt). mat (51.2MB) and out (51.2MB) both live in the 192MB L2.

#define FEAT_D 128
#define BATCH 8           // rows per TDM gather (32-bit index mode max)
#define WAVES_PER_BLOCK 8 // 256 threads / wave32

typedef __attribute__((ext_vector_type(4))) unsigned int v4u;
typedef __attribute__((ext_vector_type(8))) int v8i;
typedef __attribute__((ext_vector_type(4))) int v4i;

// Guaranteed-HW no-return float atomic adds with immediate byte offsets.
__device__ __forceinline__ void gatomic_add4_f32(float* p, float4 v) {
  asm volatile("global_atomic_add_f32 %0, %1, off"           :: "v"(p), "v"(v.x) : "memory");
  asm volatile("global_atomic_add_f32 %0, %1, off offset:4"  :: "v"(p), "v"(v.y) : "memory");
  asm volatile("global_atomic_add_f32 %0, %1, off offset:8"  :: "v"(p), "v"(v.z) : "memory");
  asm volatile("global_atomic_add_f32 %0, %1, off offset:12" :: "v"(p), "v"(v.w) : "memory");
}

// Issue one TDM gather of 8 rows (each 128 f32 = 512B) of `mat` into LDS.
// cvec holds col indices in lanes 0..7 (replicated in upper lanes).
__device__ __forceinline__ void tdm_gather8(unsigned lds_addr, const float* mat,
                                            unsigned tensor_rows, int cvec) {
  unsigned long long ga = (unsigned long long)(size_t)mat;
  v4u g0;
  g0.x = 0xC0000001u;                                   // count=1 | idx32 | gather_mode
  g0.y = lds_addr;                                      // LDS byte address of tile
  g0.z = (unsigned)(ga & 0xFFFFFFFFu);                  // global_addr[31:0]
  g0.w = (unsigned)((ga >> 32) & 0x01FFFFFFu) | (2u << 30);  // global_addr[56:32] | type=2
  v8i g1;
  g1[0] = (int)(2u << 16);                              // data_size = 4B
  g1[1] = (int)((unsigned)FEAT_D << 16);                // tensor_dim0[15:0] = 128
  g1[2] = (int)((tensor_rows & 0xFFFFu) << 16);         // dim0[31:16]=0 | tensor_dim1[15:0]
  g1[3] = (int)(((unsigned)FEAT_D << 16) | ((tensor_rows >> 16) & 0xFFFFu)); // tile_dim0 | dim1 hi
  g1[4] = BATCH;                                        // tile_dim1 = #valid indices
  g1[5] = FEAT_D;                                       // tensor_dim0_stride lo32 = 128
  g1[6] = 0;
  g1[7] = 0;
  v4i g2, g3;
  g2[0] = __builtin_amdgcn_readlane(cvec, 0);
  g2[1] = __builtin_amdgcn_readlane(cvec, 1);
  g2[2] = __builtin_amdgcn_readlane(cvec, 2);
  g2[3] = __builtin_amdgcn_readlane(cvec, 3);
  g3[0] = __builtin_amdgcn_readlane(cvec, 4);
  g3[1] = __builtin_amdgcn_readlane(cvec, 5);
  g3[2] = __builtin_amdgcn_readlane(cvec, 6);
  g3[3] = __builtin_amdgcn_readlane(cvec, 7);
  v8i gpad = {0, 0, 0, 0, 0, 0, 0, 0};                  // 6-arg toolchain extra operand
  __builtin_amdgcn_tensor_load_to_lds(g0, g1, g2, g3, gpad, 0);
}

__global__ __launch_bounds__(256) void zero_f32_kernel(float* __restrict__ out, int n) {
  int i4 = (blockIdx.x * blockDim.x + threadIdx.x) * 4;
  if (i4 + 3 < n) {
    *(float4*)(out + i4) = make_float4(0.f, 0.f, 0.f, 0.f);
  } else {
    for (int k = i4; k < n; ++k) out[k] = 0.f;
  }
}

__global__ __launch_bounds__(256) void coo_tdm_scatter_kernel(
    const float* __restrict__ mat,
    const int*   __restrict__ row,
    const int*   __restrict__ col,
    float*       __restrict__ out,
    int nnz, unsigned tensor_rows)
{
  // Per-wave double buffer: 2 x 8 rows x 128 f32 = 8KB; 64KB per 8-wave block.
  __shared__ float lds_buf[WAVES_PER_BLOCK][2][BATCH * FEAT_D];

  const int lane   = threadIdx.x & 31;
  const int waveIb = threadIdx.x >> 5;                       // wave in block
  const int wave   = (int)((blockIdx.x * blockDim.x + threadIdx.x) >> 5);
  const int W      = (int)((gridDim.x * blockDim.x) >> 5);

  const int nbatches = nnz / BATCH;

  unsigned lds_a[2];
  lds_a[0] = (unsigned)(size_t)&lds_buf[waveIb][0][0];
  lds_a[1] = (unsigned)(size_t)&lds_buf[waveIb][1][0];

  // ---- TDM-pipelined main loop over full batches of 8 nonzeros ----
  int b = wave;
  bool have = b < nbatches;
  int cvec = 0, rvec = 0;
  if (have) {
    int nz0 = b * BATCH;
    cvec = col[nz0 + (lane & 7)];
    rvec = row[nz0 + (lane & 7)];
    tdm_gather8(lds_a[0], mat, tensor_rows, cvec);
  }
  int buf = 0;
  while (have) {
    int bn = b + W;
    bool have_n = bn < nbatches;
    int cnext = 0, rnext = 0;
    if (have_n) {
      int nz0 = bn * BATCH;
      cnext = col[nz0 + (lane & 7)];
      rnext = row[nz0 + (lane & 7)];
      tdm_gather8(lds_a[buf ^ 1], mat, tensor_rows, cnext);
      asm volatile("s_wait_tensorcnt 0x1" ::: "memory");   // older gather complete
    } else {
      asm volatile("s_wait_tensorcnt 0x0" ::: "memory");   // drain all
    }

    const float* lbase = &lds_buf[waveIb][buf][0];
#pragma unroll
    for (int i = 0; i < BATCH; ++i) {
      int r = __builtin_amdgcn_readlane(rvec, i);
      float4 v = *(const float4*)(lbase + i * FEAT_D + lane * 4);   // ds_load_b128
      gatomic_add4_f32(out + (size_t)r * FEAT_D + lane * 4, v);
    }

    b = bn; cvec = cnext; rvec = rnext; buf ^= 1; have = have_n;
  }

  // ---- Tail (< BATCH leftovers): direct gather + atomics ----
  for (int nz = nbatches * BATCH + wave; nz < nnz; nz += W) {
    int c = col[nz];
    int r = row[nz];
    float4 v = *(const float4*)(mat + (size_t)c * FEAT_D + lane * 4);
    gatomic_add4_f32(out + (size_t)r * FEAT_D + lane * 4, v);
  }
}

extern "C" void kernel_launch(void* const* d_in, const int* in_sizes, int n_in,
                              void* d_out, int out_size, void* d_ws, size_t ws_size,
                              hipStream_t stream) {
  const float* mat = (const float*)d_in[0];   // [NT, 128] f32
  const int*   row = (const int*)d_in[1];     // [NNZ] i32
  const int*   col = (const int*)d_in[2];     // [NNZ] i32
  float*       out = (float*)d_out;           // [NC, 128] f32
  const int nnz = in_sizes[1];
  const unsigned tensor_rows = (unsigned)(in_sizes[0] / FEAT_D);  // NT

  // 1) Zero the (poisoned) output.
  {
    int n4 = (out_size + 3) / 4;
    int blocks = (n4 + 255) / 256;
    zero_f32_kernel<<<blocks, 256, 0, stream>>>(out, out_size);
  }

  // 2) TDM-gather + atomic scatter. 2048 blocks x 8 waves = 16384 waves,
  //    ~5 batches (40 nonzeros) per wave; two TDM descriptors in flight/wave.
  {
    int blocks = 2048;
    coo_tdm_scatter_kernel<<<blocks, 256, 0, stream>>>(mat, row, col, out,
                                                       nnz, tensor_rows);
  }
}